// OctahedralSHT_7799660609693
// MI455X (gfx1250) — compile-verified
//
#include <hip/hip_runtime.h>

// ---------------------------------------------------------------------------
// Octahedral SHT on MI455X (gfx1250).
// Two batched-GEMM stages mapped onto V_WMMA_F32_16X16X4_F32 (full f32
// precision, matching the f32 reference). Problem is memory-bound
// (~6.6 GFLOP vs ~150 MB traffic), so f32 WMMA is the right precision.
// ---------------------------------------------------------------------------

typedef __attribute__((ext_vector_type(2))) float v2f;
typedef __attribute__((ext_vector_type(4))) float v4f;
typedef __attribute__((ext_vector_type(8))) float v8f;

#define NLAT 192
#define MAXJ 400
#define MM   128      // mmax
#define LM   128      // lmax
#define NPTS 40320    // sum of nlon over octahedral grid
#define RTOT 128      // b*e*v = 2*1*64
#define VDIM 64

// workspace layout (in floats)
#define PWT_SZ  ((size_t)MM * NLAT * LM)    // 3,145,728 floats (12.6 MB)
#define FM_HALF ((size_t)MM * NLAT * RTOT)  // 3,145,728 floats per re/im

static __device__ __forceinline__ v8f wmma_f32x4(v2f a, v2f b, v8f c) {
    return __builtin_amdgcn_wmma_f32_16x16x4_f32(
        /*neg_a=*/false, a, /*neg_b=*/false, b,
        /*c_mod=*/(short)0, c, /*reuse_a=*/false, /*reuse_b=*/false);
}

// ---------------------------------------------------------------------------
// Kernel 0: Pw[l][m][n] -> PwT[m][n][l]  (coalesced writes, one-time 12.6 MB)
// ---------------------------------------------------------------------------
__global__ __launch_bounds__(256) void sht_transpose_pw(
    const float* __restrict__ Pw, float* __restrict__ PwT)
{
    int idx = blockIdx.x * 256 + threadIdx.x;
    if (idx >= (int)PWT_SZ) return;
    int l = idx & 127;
    int t = idx >> 7;
    int n = t % NLAT;
    int m = t / NLAT;
    PwT[idx] = Pw[((size_t)l * MM + m) * NLAT + n];
}

// ---------------------------------------------------------------------------
// Kernel 1: Fourier stage.
//   Fm_{re,im}[m][n][r] = sum_j data[b, pad_idx[n,j], v] * E_{re,im}[n][j][m]
// grid = (192 latitudes, 2 b-blocks), block = 256 (8 waves).
// Wave w owns m-columns [16w,16w+16); accumulates 4 M-tiles x {re,im}.
// A panel (gathered, 64 rows x 16 k) staged in LDS, k-pair interleaved so an
// A fragment is one ds_load_b64 per tile.
// ---------------------------------------------------------------------------
__global__ __launch_bounds__(256) void sht_fourier(
    const float* __restrict__ data,  const float* __restrict__ E_re,
    const float* __restrict__ E_im,  const int* __restrict__ pad_idx,
    float* __restrict__ Fm)
{
    // A2[pair p][row r][c] = A[row r][k = 2p+c], chunk of K=16 -> 8 pairs
    __shared__ __align__(16) float A2[8 * 64 * 2];

    const int n   = blockIdx.x;         // latitude
    const int mb  = blockIdx.y;         // 0..1 == batch b (r = b*64 + v)
    const int tid = threadIdx.x;
    const int w   = tid >> 5;           // wave 0..7
    const int L   = tid & 31;
    const int lh  = L >> 4;             // lane half (selects K pair)
    const int ll  = L & 15;
    const int mcol = w * 16 + ll;       // this lane's B column (m)

    const v8f vzero = {0.f,0.f,0.f,0.f,0.f,0.f,0.f,0.f};
    v8f acc_re[4], acc_im[4];
#pragma unroll
    for (int t = 0; t < 4; ++t) { acc_re[t] = vzero; acc_im[t] = vzero; }

    const size_t ebase = (size_t)n * MAXJ * MM;

    for (int kk = 0; kk < MAXJ; kk += 16) {   // 400/16 = 25 chunks
        __syncthreads();
        // cooperative gather of A chunk: 16 j-cols x 64 rows, coalesced in v
#pragma unroll
        for (int i = 0; i < 4; ++i) {
            int idx = tid + i * 256;
            int j   = idx >> 6;           // 0..15
            int rv  = idx & 63;           // v within this b-block
            int p   = pad_idx[n * MAXJ + kk + j];
            float val = data[((size_t)mb * NPTS + p) * VDIM + rv];
            A2[((j >> 1) * 64 + rv) * 2 + (j & 1)] = val;
        }
        __syncthreads();

#pragma unroll
        for (int ks = 0; ks < 4; ++ks) {  // 4 K-steps of 4 per chunk
            const int k0 = ks * 4;
            const int jg = kk + k0 + 2 * lh;
            v2f bre, bim;
            bre.x = E_re[ebase + (size_t)jg       * MM + mcol];
            bre.y = E_re[ebase + (size_t)(jg + 1) * MM + mcol];
            bim.x = E_im[ebase + (size_t)jg       * MM + mcol];
            bim.y = E_im[ebase + (size_t)(jg + 1) * MM + mcol];
            const int p = (k0 >> 1) + lh;
#pragma unroll
            for (int t = 0; t < 4; ++t) {
                v2f a = *(const v2f*)&A2[(p * 64 + t * 16 + ll) * 2];
                acc_re[t] = wmma_f32x4(a, bre, acc_re[t]);
                acc_im[t] = wmma_f32x4(a, bim, acc_im[t]);
            }
        }
    }

    // store Fm[reim][m][n][r] (contiguous-in-(n,r) slabs for stage 2)
#pragma unroll
    for (int t = 0; t < 4; ++t) {
#pragma unroll
        for (int v = 0; v < 8; ++v) {
            int r = mb * 64 + t * 16 + v + lh * 8;
            size_t o = ((size_t)mcol * NLAT + n) * RTOT + r;
            Fm[o]           = acc_re[t][v];
            Fm[FM_HALF + o] = acc_im[t][v];
        }
    }
}

// ---------------------------------------------------------------------------
// Kernel 2: Legendre stage.
//   coeffs[r][l] = sum_n Fm[m][n][r] * PwT[m][n][l]     (per m, per re/im)
// grid = (128 m, 2 reim), block = 256 (8 waves).
// Wave w owns l-tile w and all 8 M-tiles (r = 0..127).
// A slab Fm[m] is contiguous; chunked through LDS with float4 loads.
// ---------------------------------------------------------------------------
__global__ __launch_bounds__(256) void sht_legendre(
    const float* __restrict__ Fm, const float* __restrict__ PwT,
    float* __restrict__ out)
{
    __shared__ __align__(16) float A2[8 * 128 * 2]; // [pair][r][c]

    const int m    = blockIdx.x;
    const int reim = blockIdx.y;
    const int tid  = threadIdx.x;
    const int w    = tid >> 5;
    const int L    = tid & 31;
    const int lh   = L >> 4;
    const int ll   = L & 15;
    const int lcol = w * 16 + ll;       // output l column

    const float* __restrict__ Abase = Fm  + (size_t)reim * FM_HALF
                                          + (size_t)m * NLAT * RTOT;
    const float* __restrict__ Bbase = PwT + (size_t)m * NLAT * LM;

    const v8f vzero = {0.f,0.f,0.f,0.f,0.f,0.f,0.f,0.f};
    v8f acc[8];
#pragma unroll
    for (int t = 0; t < 8; ++t) acc[t] = vzero;

    for (int kk = 0; kk < NLAT; kk += 16) {   // 192/16 = 12 chunks
        __syncthreads();
        // chunk = 16 n-rows x 128 r = 2048 floats, fully contiguous in memory
#pragma unroll
        for (int i = 0; i < 2; ++i) {
            int idx4 = tid + i * 256;         // 512 float4 total
            int nl   = idx4 >> 5;             // 0..15 (n within chunk)
            int r    = (idx4 & 31) * 4;
            v4f val = *(const v4f*)&Abase[(size_t)(kk + nl) * RTOT + r];
            int p = nl >> 1, c = nl & 1;
#pragma unroll
            for (int q = 0; q < 4; ++q)
                A2[(p * 128 + r + q) * 2 + c] = val[q];
        }
        __syncthreads();

#pragma unroll
        for (int ks = 0; ks < 4; ++ks) {
            const int k0 = ks * 4;
            const int ng = kk + k0 + 2 * lh;
            v2f b;
            b.x = Bbase[(size_t)ng       * LM + lcol];
            b.y = Bbase[(size_t)(ng + 1) * LM + lcol];
            const int p = (k0 >> 1) + lh;
#pragma unroll
            for (int t = 0; t < 8; ++t) {
                v2f a = *(const v2f*)&A2[(p * 128 + t * 16 + ll) * 2];
                acc[t] = wmma_f32x4(a, b, acc[t]);
            }
        }
    }

    // out is complex64 viewed as float pairs: [b][e=1][l][m][v][{re,im}]
#pragma unroll
    for (int t = 0; t < 8; ++t) {
#pragma unroll
        for (int v = 0; v < 8; ++v) {
            int r  = t * 16 + v + lh * 8;     // r = b*64 + vv
            int b  = r >> 6;
            int vv = r & 63;
            size_t o = ((((size_t)b * LM + lcol) * MM + m) * VDIM + vv) * 2 + reim;
            out[o] = acc[t][v];
        }
    }
}

// ---------------------------------------------------------------------------
// Launch: transpose Pw, Fourier stage, Legendre stage (all on `stream`).
// ws usage: PwT (12.6 MB) + Fm re/im (25.2 MB) = 37.8 MB.
// ---------------------------------------------------------------------------
extern "C" void kernel_launch(void* const* d_in, const int* in_sizes, int n_in,
                              void* d_out, int out_size, void* d_ws, size_t ws_size,
                              hipStream_t stream) {
    const float* data  = (const float*)d_in[0];   // [2,1,40320,64] f32
    const float* Pw    = (const float*)d_in[1];   // [128,128,192]  f32
    const float* E_re  = (const float*)d_in[2];   // [192,400,128]  f32
    const float* E_im  = (const float*)d_in[3];   // [192,400,128]  f32
    const int*   padix = (const int*)  d_in[4];   // [192,400]      i32

    float* ws  = (float*)d_ws;
    float* PwT = ws;
    float* Fm  = ws + PWT_SZ;
    float* out = (float*)d_out;                   // complex64 -> float pairs

    (void)in_sizes; (void)n_in; (void)out_size; (void)ws_size;

    int tot = (int)PWT_SZ;
    sht_transpose_pw<<<(tot + 255) / 256, 256, 0, stream>>>(Pw, PwT);

    dim3 g1(NLAT, 2);
    sht_fourier<<<g1, 256, 0, stream>>>(data, E_re, E_im, padix, Fm);

    dim3 g2(MM, 2);
    sht_legendre<<<g2, 256, 0, stream>>>(Fm, PwT, out);
}